// ParallelEmbedding_11295763988601
// MI455X (gfx1250) — compile-verified
//
#include <hip/hip_runtime.h>

// Reference constants: P=8, D=1024. V is derived from in_sizes at launch.
constexpr int   kP   = 8;
constexpr int   kD   = 1024;
constexpr float kEps = 0.01f;

typedef float v4f __attribute__((ext_vector_type(4)));

__global__ __launch_bounds__(256) void pembed_noise_kernel(
    const int*   __restrict__ ids,     // [n_tokens]        (B*S)
    const float* __restrict__ weight,  // [V, D]
    const float* __restrict__ mu,      // [P, V, D]
    float*       __restrict__ out,     // [P, n_tokens, D]
    int n_tokens, int vocab)
{
    // 32 KB staging buffer: the 8 mu rows for this token.
    __shared__ v4f smem[kP * (kD / 4)];

    const int token = blockIdx.x;                 // one token per workgroup
    const int t     = threadIdx.x;                // 0..255 -> one float4 of the row
    const long long id = (long long)ids[token];   // block-uniform -> s_load_b32

    // Weight row: issue first (everything depends on it), stays in VGPRs, reused x8.
    v4f w = *(const v4f*)(weight + (size_t)id * kD + (size_t)(t * 4));

    const size_t   mu_pstride = (size_t)vocab * kD;      // stride between noise copies
    const float*   mu_row     = mu + (size_t)id * kD;    // p=0 row base (SGPR pair)
    const uint32_t lane_byte  = (uint32_t)(t * 16u);     // per-lane byte offset in a row

    // Stage all 8 mu rows into LDS with CDNA5 async copies (tracked by ASYNCcnt).
    // GVS addressing: SGPR row base + 32-bit per-lane VGPR byte offset.
    // LDS dest address = low 32 bits of the generic shared pointer
    // (flat LDS aperture: addr[31:0] is the LDS byte offset).
#pragma unroll
    for (int p = 0; p < kP; ++p) {
        const float* src = mu_row + (size_t)p * mu_pstride;           // uniform -> SGPR pair
        uint32_t lds_addr = (uint32_t)(size_t)(&smem[p * (kD / 4) + t]);
        asm volatile("global_load_async_to_lds_b128 %0, %1, %2"
                     :
                     : "v"(lds_addr), "v"(lane_byte), "s"(src)
                     : "memory");
    }

    const v4f eps = {kEps, kEps, kEps, kEps};
    const size_t out_base    = (size_t)token * kD + (size_t)(t * 4);
    const size_t out_pstride = (size_t)n_tokens * kD;

    // Incremental drain: async loads complete in order within a wave, so row p is
    // resident in LDS once ASYNCcnt <= (7-p). Compute/NT-store of row p overlaps
    // the remaining rows still in flight.
#define PEMBED_STEP(p)                                                                 \
    {                                                                                  \
        asm volatile("s_wait_asynccnt %0" : : "n"(kP - 1 - (p)) : "memory");           \
        v4f m = smem[(p) * (kD / 4) + t];   /* ds_load_b128 */                         \
        v4f o = w + eps * m;                /* packed FMAs  */                         \
        /* 268 MB write-once stream: non-temporal so it doesn't thrash L2, which   */ \
        /* we want holding the mu/weight rows (duplicate-token reuse).             */ \
        __builtin_nontemporal_store(                                                   \
            o, (v4f*)(out + (size_t)(p) * out_pstride + out_base));                    \
    }

    PEMBED_STEP(0)
    PEMBED_STEP(1)
    PEMBED_STEP(2)
    PEMBED_STEP(3)
    PEMBED_STEP(4)
    PEMBED_STEP(5)
    PEMBED_STEP(6)
    PEMBED_STEP(7)
#undef PEMBED_STEP
}

extern "C" void kernel_launch(void* const* d_in, const int* in_sizes, int n_in,
                              void* d_out, int out_size, void* d_ws, size_t ws_size,
                              hipStream_t stream) {
    const int*   ids    = (const int*)  d_in[0];   // input_ids (harness: integer -> int*)
    const float* weight = (const float*)d_in[1];   // [V, D] f32
    const float* mu     = (const float*)d_in[2];   // [P, V, D] f32
    float*       out    = (float*)d_out;           // [P, B*S, D] f32

    const int n_tokens = in_sizes[0];              // B*S = 8192
    const int vocab    = in_sizes[1] / kD;         // 50257

    pembed_noise_kernel<<<dim3(n_tokens), dim3(256), 0, stream>>>(
        ids, weight, mu, out, n_tokens, vocab);
}